// AttNLocal_15736760172586
// MI455X (gfx1250) — compile-verified
//
#include <hip/hip_runtime.h>
#include <stdint.h>

// Problem constants (match the reference).
#define LEN    4096   // L == maxlen (rows and cols of x)
#define LIMIT  256    // band width
#define BATCH  4
#define ROWS   8      // rows gathered per block
#define TPB    256    // 8 wave32 waves per block

// out[b,i,j] = x[b,i,i+j] if i+j < LEN else 0.
//
// Phase 1: per-lane dword async gather of the (row-misaligned) diagonal band
//          into LDS via the gfx1250 async global->LDS path (ASYNCcnt).
// Phase 2: aligned 128-bit stores of the contiguous output rows.
//
// EDGE=false: block is fully interior (no lane can run past the tensor edge)
//             -> no EXEC predication, no zero masking.
// EDGE=true : per-lane predicated loads + per-element zero masking.
template <bool EDGE>
__device__ __forceinline__ void band_tile(const float* __restrict__ x,
                                          float* __restrict__ out,
                                          float* tile, int tid, int gr0) {
    // ---- Phase 1: async global -> LDS gather ----
#pragma unroll
    for (int r = 0; r < ROWS; ++r) {
        const int gr = gr0 + r;
        const int i  = gr & (LEN - 1);     // row within the batch image
        const int j  = tid;                // band column, one dword per lane
        // flat element offset: b*LEN*LEN + i*LEN + (i+j) == gr*LEN + i + j
        const unsigned long long gaddr =
            (unsigned long long)(x + ((unsigned long long)gr * LEN + i + j));
        const unsigned lds_addr =
            (unsigned)(uintptr_t)(&tile[r * LIMIT + j]);   // low 32 bits == LDS offset
        if (!EDGE || (i + j < LEN)) {      // lane predicate only on edge blocks
            asm volatile("global_load_async_to_lds_b32 %0, %1, off"
                         :
                         : "v"(lds_addr), "v"(gaddr)
                         : "memory");
        }
    }

    // Drain this wave's async copies, then make the tile visible across waves.
    asm volatile("s_wait_asynccnt 0" ::: "memory");
    __syncthreads();

    // ---- Phase 2: aligned float4 write-out ----
#pragma unroll
    for (int it = 0; it < (ROWS * LIMIT) / (4 * TPB); ++it) {   // 2 iterations
        const int q  = it * TPB + tid;     // float4 slot within the tile
        const int r  = q >> 6;             // LIMIT/4 == 64 float4 per row
        const int j  = (q & 63) * 4;       // element column
        const int gr = gr0 + r;

        float4 v = *reinterpret_cast<const float4*>(&tile[r * LIMIT + j]);
        if (EDGE) {
            const int i = gr & (LEN - 1);
            if (i + j + 3 >= LEN) {
                v.x = (i + j + 0 < LEN) ? v.x : 0.0f;
                v.y = (i + j + 1 < LEN) ? v.y : 0.0f;
                v.z = (i + j + 2 < LEN) ? v.z : 0.0f;
                v.w = (i + j + 3 < LEN) ? v.w : 0.0f;
            }
        }
        *reinterpret_cast<float4*>(out + ((unsigned long long)gr * LIMIT + j)) = v;
    }
}

__global__ __launch_bounds__(TPB) void band_gather_kernel(const float* __restrict__ x,
                                                          float* __restrict__ out) {
    __shared__ float tile[ROWS * LIMIT];   // 8 KB of the WGP's 320 KB LDS

    const int tid = threadIdx.x;
    const int gr0 = blockIdx.x * ROWS;     // first flat row index (b*LEN + i)
    const int i0  = gr0 & (LEN - 1);       // first row within the batch image

    // Block is interior iff even its last row's last band column is in range:
    // (i0 + ROWS-1) + (LIMIT-1) < LEN. Uniform in blockIdx -> scalar branch.
    if (i0 + ROWS + LIMIT - 2 < LEN) {
        band_tile<false>(x, out, tile, tid, gr0);
    } else {
        band_tile<true>(x, out, tile, tid, gr0);
    }
}

extern "C" void kernel_launch(void* const* d_in, const int* in_sizes, int n_in,
                              void* d_out, int out_size, void* d_ws, size_t ws_size,
                              hipStream_t stream) {
    (void)in_sizes; (void)n_in; (void)out_size; (void)d_ws; (void)ws_size;
    const float* x = (const float*)d_in[0];
    float* out     = (float*)d_out;

    const int total_rows = BATCH * LEN;            // 16384
    dim3 grid(total_rows / ROWS);                  // 2048 blocks
    dim3 block(TPB);                               // 256 threads = 8 waves
    band_gather_kernel<<<grid, block, 0, stream>>>(x, out);
}